// VectorQuantizer_76845554860663
// MI455X (gfx1250) — compile-verified
//
#include <hip/hip_runtime.h>

typedef __attribute__((ext_vector_type(2))) float v2f;
typedef __attribute__((ext_vector_type(8))) float v8f;

#define EMBED_DIM 64
#define WAVES_PER_BLOCK 4
#define BLOCK_THREADS 128

// ||c_k||^2 for each codebook row -> d_ws
__global__ void vq_cnorm_kernel(const float* __restrict__ C,
                                float* __restrict__ cnorm, int K) {
  int k = blockIdx.x * blockDim.x + threadIdx.x;
  if (k >= K) return;
  const float* row = C + (long)k * EMBED_DIM;
  float s = 0.f;
#pragma unroll
  for (int i = 0; i < EMBED_DIM / 4; ++i) {
    float4 v = *(const float4*)(row + 4 * i);
    s += v.x * v.x + v.y * v.y + v.z * v.z + v.w * v.w;
  }
  cnorm[k] = s;
}

// One workgroup = 16 input rows. 4 waves split the codebook (K/4 entries each),
// each wave runs f32 WMMA 16x16x4 chains over 16-column codebook chunks,
// tracks per-lane running (min score, argmin), then cross-lane + cross-wave
// reduction, gather + elementwise outputs.
__global__ __launch_bounds__(BLOCK_THREADS) void vq_main_kernel(
    const float* __restrict__ X, const float* __restrict__ C,
    const float* __restrict__ cnorm, float* __restrict__ loss,
    float* __restrict__ qout, int N, int K) {
  const int tid = threadIdx.x;
  const int wave = tid >> 5;
  const int lane = tid & 31;
  const int r0 = blockIdx.x * 16;
  const int nl = lane & 15;   // row (A) / column (B,C) index within tile
  const int half = lane >> 4; // K-half selector for A/B fragments

  // A fragments: element (m, k=4s+2*half+j) in lane (half*16+m), vgpr j.
  v2f a[16];
  {
    const float* xrow = X + (long)(r0 + nl) * EMBED_DIM + 2 * half;
#pragma unroll
    for (int s = 0; s < 16; ++s) a[s] = *(const v2f*)(xrow + 4 * s);
  }

  float bestS[8];
  int bestI[8];
#pragma unroll
  for (int j = 0; j < 8; ++j) {
    bestS[j] = __builtin_inff();
    bestI[j] = 0x7fffffff;
  }

  const int kPerWave = K / WAVES_PER_BLOCK;
  const int kBegin = wave * kPerWave;
  const int kEnd = kBegin + kPerWave;
  for (int c0 = kBegin; c0 < kEnd; c0 += 16) {
    // B fragments: element (k=4s+2*half+j, n) in lane (half*16+n), vgpr j.
    const float* brow = C + (long)(c0 + nl) * EMBED_DIM + 2 * half;
    v8f acc = {0.f, 0.f, 0.f, 0.f, 0.f, 0.f, 0.f, 0.f};
#pragma unroll
    for (int s = 0; s < 16; ++s) {
      v2f b = *(const v2f*)(brow + 4 * s);
      acc = __builtin_amdgcn_wmma_f32_16x16x4_f32(
          /*neg_a=*/false, a[s], /*neg_b=*/false, b,
          /*c_mod=*/(short)0, acc, /*reuse_a=*/false, /*reuse_b=*/false);
    }
    // C/D layout: lane half h, vgpr j -> row m = 8h+j, col n = lane&15.
    const float cn = cnorm[c0 + nl];
    const int myIdx = c0 + nl;
#pragma unroll
    for (int j = 0; j < 8; ++j) {
      float sc = __builtin_fmaf(-2.f, acc[j], cn);  // ||c||^2 - 2 x.c
      if (sc < bestS[j]) { bestS[j] = sc; bestI[j] = myIdx; }  // idx monotone -> first-min
    }
  }

  // Reduce over the 16 column-lanes (xor masks < 16 stay within each half).
#pragma unroll
  for (int m = 1; m < 16; m <<= 1) {
#pragma unroll
    for (int j = 0; j < 8; ++j) {
      float os = __shfl_xor(bestS[j], m, 32);
      int oi = __shfl_xor(bestI[j], m, 32);
      if (os < bestS[j] || (os == bestS[j] && oi < bestI[j])) {
        bestS[j] = os;
        bestI[j] = oi;
      }
    }
  }

  __shared__ float sS[WAVES_PER_BLOCK][16];
  __shared__ int sI[WAVES_PER_BLOCK][16];
  __shared__ int finalIdx[16];
  if (nl == 0) {
#pragma unroll
    for (int j = 0; j < 8; ++j) {
      sS[wave][half * 8 + j] = bestS[j];
      sI[wave][half * 8 + j] = bestI[j];
    }
  }
  __syncthreads();
  if (tid < 16) {
    float bs = sS[0][tid];
    int bi = sI[0][tid];
#pragma unroll
    for (int w = 1; w < WAVES_PER_BLOCK; ++w) {
      float s2 = sS[w][tid];
      int i2 = sI[w][tid];
      if (s2 < bs || (s2 == bs && i2 < bi)) { bs = s2; bi = i2; }
    }
    finalIdx[tid] = bi;
  }
  __syncthreads();

  // Gather + elementwise outputs (match reference rounding exactly).
  for (int e = tid; e < 16 * EMBED_DIM; e += BLOCK_THREADS) {
    int row = e >> 6;
    int d = e & 63;
    long gi = (long)(r0 + row) * EMBED_DIM + d;
    float q = C[(long)finalIdx[row] * EMBED_DIM + d];
    float x = X[gi];
    float diff = q - x;                 // sg(q) - x == q - sg(x)
    float d2v = diff * diff;
    loss[gi] = d2v + 0.25f * d2v;       // q_latent + 0.25 * e_latent
    qout[gi] = x + diff;                // straight-through
  }
}

extern "C" void kernel_launch(void* const* d_in, const int* in_sizes, int n_in,
                              void* d_out, int out_size, void* d_ws,
                              size_t ws_size, hipStream_t stream) {
  const float* X = (const float*)d_in[0];  // [16,32,32,64] f32
  const float* C = (const float*)d_in[1];  // [8192,64] f32
  int N = in_sizes[0] / EMBED_DIM;         // 16384
  int K = in_sizes[1] / EMBED_DIM;         // 8192
  float* cnorm = (float*)d_ws;             // K floats of scratch
  float* loss = (float*)d_out;             // outputs concatenated: loss, then quantized_st
  float* qout = loss + (long)N * EMBED_DIM;

  vq_cnorm_kernel<<<(K + 255) / 256, 256, 0, stream>>>(C, cnorm, K);
  vq_main_kernel<<<N / 16, BLOCK_THREADS, 0, stream>>>(X, C, cnorm, loss, qout,
                                                       N, K);
}